// ExpLeak_59674275611337
// MI455X (gfx1250) — compile-verified
//
#include <hip/hip_runtime.h>
#include <hip/hip_bf16.h>
#include <cstdint>

// ExpLeak streaming scan for MI455X (gfx1250).
// Memory-bound (0.25 FLOP/B, ~512 MiB traffic -> ~23 us roofline): stream
// [B,T,N] once in / once out through a per-wave async-to-LDS ring pipeline
// (ASYNCcnt), ds_load_b128 consume, non-temporal b128 stores, and
// global_prefetch to warm L2 ahead of the ASYNCcnt window.

typedef float v4f __attribute__((ext_vector_type(4)));

namespace {
constexpr int kT = 1024;
constexpr int kN = 4096;
constexpr int kB = 16;

constexpr int kDepth     = 32;               // LDS ring slots (t-steps) per wave
constexpr int kU         = 4;                // t-steps consumed per iteration
constexpr int kWaves     = 4;                // waves per block
constexpr int kThreads   = 32 * kWaves;      // 128
constexpr int kVec       = 4;                // floats per lane (b128)
constexpr int kNPerBlock = kThreads * kVec;  // 512 n-columns per block
constexpr int kSlotBytes = 32 * kVec * 4;    // 512 B per wave per slot
}  // namespace

// global -> LDS async copy, 16B per lane. GVS addressing:
//   global address is the 64-bit base in sbase plus the 32-bit offset in voff; LDS dest is in lds_off.
__device__ __forceinline__ void async_copy_b128(unsigned lds_off, unsigned voff,
                                                const float* sbase) {
  asm volatile("global_load_async_to_lds_b128 %0, %1, %2"
               :: "v"(lds_off), "v"(voff), "s"(sbase)
               : "memory");
}

template <int N>
__device__ __forceinline__ void wait_asynccnt() {
  asm volatile("s_wait_asynccnt %0" :: "i"(N) : "memory");
}

__device__ __forceinline__ void wait_dscnt0() {
  asm volatile("s_wait_dscnt 0" ::: "memory");
}

__global__ __launch_bounds__(kThreads)
void expleak_scan_async(const float* __restrict__ in,
                        const float* __restrict__ tau_mem,
                        float* __restrict__ out) {
  __shared__ __align__(16) float ring[kWaves][kDepth][32 * kVec];

  const int tid  = threadIdx.x;
  const int wv   = tid >> 5;
  const int lane = tid & 31;
  const int bpb  = kN / kNPerBlock;            // blocks per batch row (8)
  const int b    = blockIdx.x / bpb;
  const int nb   = blockIdx.x % bpb;
  const int n0   = nb * kNPerBlock + tid * kVec;

  const float alpha = __expf(-1.0f / tau_mem[0]);

  const float* gbase = in + (size_t)b * kT * kN;        // uniform per block
  const unsigned vstep = (unsigned)kN * 4u;             // 16 KiB per t-row
  unsigned pvoff = (unsigned)n0 * 4u;                   // issue cursor (t=0)
  // Low 32 bits of a generic pointer to __shared__ == LDS byte offset.
  const unsigned lds0 = (unsigned)(uintptr_t)&ring[wv][0][lane * kVec];

  // ---- prologue: fill the async ring with t = 0 .. kDepth-1 ----
  #pragma unroll
  for (int d = 0; d < kDepth; ++d) {
    async_copy_b128(lds0 + (unsigned)(d * kSlotBytes), pvoff, gbase);
    pvoff += vstep;
  }

  v4f st = {0.f, 0.f, 0.f, 0.f};
  float* optr = out + (size_t)b * kT * kN + n0;
  const float* pref = gbase + n0;

  int slot = 0;
  // ---- main pipelined loop: consume kU steps, refill kU slots ----
  for (int t = 0; t < kT - kDepth; t += kU) {
    wait_asynccnt<kDepth - kU>();   // oldest kU slots have landed in LDS
    #pragma unroll
    for (int i = 0; i < kU; ++i) {
      const v4f x = *(const v4f*)&ring[wv][slot + i][lane * kVec];
      st = alpha * st + x;
      __builtin_nontemporal_store(st, (v4f*)optr);
      optr += kN;
    }
    wait_dscnt0();                  // our ds reads done before slots are reused
    #pragma unroll
    for (int i = 0; i < kU; ++i) {
      async_copy_b128(lds0 + (unsigned)((slot + i) * kSlotBytes), pvoff, gbase);
      pvoff += vstep;
    }
    if (t + 2 * kDepth + kU <= kT) {  // warm L2 beyond the ASYNCcnt window
      #pragma unroll
      for (int i = 0; i < kU; ++i)
        __builtin_prefetch(pref + (size_t)(t + 2 * kDepth + i) * kN, 0, 3);
    }
    slot += kU;
    if (slot == kDepth) slot = 0;
  }

  // ---- tail: everything already issued; drain the ring ----
  wait_asynccnt<0>();
  for (int t = kT - kDepth; t < kT; t += kU) {
    #pragma unroll
    for (int i = 0; i < kU; ++i) {
      const v4f x = *(const v4f*)&ring[wv][slot + i][lane * kVec];
      st = alpha * st + x;
      __builtin_nontemporal_store(st, (v4f*)optr);
      optr += kN;
    }
    slot += kU;
    if (slot == kDepth) slot = 0;
  }
}

extern "C" void kernel_launch(void* const* d_in, const int* in_sizes, int n_in,
                              void* d_out, int out_size, void* d_ws, size_t ws_size,
                              hipStream_t stream) {
  (void)in_sizes; (void)n_in; (void)out_size; (void)d_ws; (void)ws_size;
  const float* in  = (const float*)d_in[0];   // input_current [B,T,N] f32
  const float* tau = (const float*)d_in[1];   // tau_mem [1] f32
  float* out = (float*)d_out;                 // [B,T,N] f32

  dim3 grid(kB * (kN / kNPerBlock));          // 16 * 8 = 128 blocks
  expleak_scan_async<<<grid, dim3(kThreads), 0, stream>>>(in, tau, out);
}